// GraphConvolution_44418551775394
// MI455X (gfx1250) — compile-verified
//
#include <hip/hip_runtime.h>

typedef __attribute__((ext_vector_type(2))) float v2f;
typedef __attribute__((ext_vector_type(8))) float v8f;

#define NN 100000   // nodes
#define FF 128      // F_IN == F_OUT

// ---------------------------------------------------------------------------
// Kernel 1: out[i][j] = bias[j]   (fuses the "+ bias" and initializes d_out,
// which the harness poisons). Each thread writes one float4.
// ---------------------------------------------------------------------------
__global__ void gcn_init_bias(float* __restrict__ out,
                              const float* __restrict__ bias) {
    long long idx = (long long)blockIdx.x * blockDim.x + threadIdx.x;
    long long base = idx * 4;
    if (base < (long long)NN * FF) {
        float4 b = *(const float4*)(bias + (int)(base & (FF - 1)));
        *(float4*)(out + base) = b;
    }
}

// ---------------------------------------------------------------------------
// Kernel 2: support = X @ W with V_WMMA_F32_16X16X4_F32.
// Block = 256 threads = 8 waves. Block b computes rows [16b,16b+16);
// wave w computes the 16x16 tile at columns [16w,16w+16). K loop = 32 WMMAs.
// A (16x4 f32): lane l -> m=l%16, k = 2*(l/16)+{0,1}  (one float2 load)
// B (4x16 f32): lane l -> n=l%16, k = 2*(l/16)+{0,1}
// D (16x16 f32, 8 VGPRs): vgpr r -> row r + 8*(l/16), col l%16
// ---------------------------------------------------------------------------
__global__ void __launch_bounds__(256)
gcn_gemm_wmma(const float* __restrict__ X,
              const float* __restrict__ W,
              float* __restrict__ S) {
    const int lane = threadIdx.x & 31;
    const int wave = threadIdx.x >> 5;        // 0..7
    const int row0 = blockIdx.x * 16;
    const int col0 = wave * 16;
    const int m    = lane & 15;               // also n for B
    const int half = lane >> 4;               // 0 or 1
    const int kb   = half * 2;

    const float* Arow = X + (size_t)(row0 + m) * FF;  // + k + kb
    const float* Bcol = W + col0 + m;                 // + (k+kb)*FF

    v8f c = {};
#pragma unroll
    for (int k = 0; k < FF; k += 4) {
        v2f a = *(const v2f*)(Arow + k + kb);
        v2f b;
        b.x = Bcol[(size_t)(k + kb) * FF];
        b.y = Bcol[(size_t)(k + kb + 1) * FF];
        // (neg_a, A, neg_b, B, c_mod, C, reuse_a, reuse_b)
        c = __builtin_amdgcn_wmma_f32_16x16x4_f32(
                false, a, false, b, (short)0, c, false, false);
    }

    float* Srow = S + (size_t)(row0 + half * 8) * FF + col0 + m;
#pragma unroll
    for (int r = 0; r < 8; ++r)
        Srow[(size_t)r * FF] = c[r];
}

// ---------------------------------------------------------------------------
// Kernel 3: scatter-add. One wave handles 4 consecutive edges; for each edge
// the 32 lanes cover the 128-float support row with float4 loads (L2-resident,
// 51.2 MB << 192 MB L2) and do 4 global_atomic_add_f32 each into out.
// Prefetch the next edge's row to overlap gather latency.
// ---------------------------------------------------------------------------
#define EDGES_PER_WAVE 4

__global__ void gcn_scatter(const int* __restrict__ erow,
                            const int* __restrict__ ecol,
                            const float* __restrict__ eval,
                            const float* __restrict__ S,
                            float* __restrict__ out,
                            int E) {
    const int lane = threadIdx.x & 31;
    const long long wave = ((long long)blockIdx.x * blockDim.x + threadIdx.x) >> 5;
    const long long e0 = wave * EDGES_PER_WAVE;

#pragma unroll
    for (int i = 0; i < EDGES_PER_WAVE; ++i) {
        long long e = e0 + i;
        if (e >= E) return;

        int   r = erow[e];
        int   cidx = ecol[e];
        float v = eval[e];

        if (i + 1 < EDGES_PER_WAVE && e + 1 < E) {
            // global_prefetch_b8 of the next edge's support row
            __builtin_prefetch(S + (size_t)ecol[e + 1] * FF + lane * 4, 0, 1);
        }

        float4 sv = *(const float4*)(S + (size_t)cidx * FF + lane * 4);
        float* o  = out + (size_t)r * FF + lane * 4;

        unsafeAtomicAdd(o + 0, v * sv.x);   // global_atomic_add_f32
        unsafeAtomicAdd(o + 1, v * sv.y);
        unsafeAtomicAdd(o + 2, v * sv.z);
        unsafeAtomicAdd(o + 3, v * sv.w);
    }
}

// ---------------------------------------------------------------------------
// Host side
// ---------------------------------------------------------------------------
extern "C" void kernel_launch(void* const* d_in, const int* in_sizes, int n_in,
                              void* d_out, int out_size, void* d_ws, size_t ws_size,
                              hipStream_t stream) {
    const float* x    = (const float*)d_in[0];   // [N, 128]
    const int*   erow = (const int*)  d_in[1];   // [E]
    const int*   ecol = (const int*)  d_in[2];   // [E]
    const float* ev   = (const float*)d_in[3];   // [E]
    const float* w    = (const float*)d_in[4];   // [128, 128]
    const float* bias = (const float*)d_in[5];   // [128]

    float* out = (float*)d_out;                  // [N, 128]
    float* S   = (float*)d_ws;                   // support scratch: N*128 floats
    const int E = in_sizes[1];

    // 1) out = bias (broadcast)
    {
        long long threads = (long long)NN * FF / 4;
        int blocks = (int)((threads + 255) / 256);
        gcn_init_bias<<<blocks, 256, 0, stream>>>(out, bias);
    }

    // 2) S = X @ W  (WMMA f32 16x16x4, 8 waves/block, 6250 blocks)
    gcn_gemm_wmma<<<NN / 16, 256, 0, stream>>>(x, w, S);

    // 3) out[row] += val * S[col]  (atomic scatter)
    {
        long long waves = (E + EDGES_PER_WAVE - 1) / EDGES_PER_WAVE;
        long long threads = waves * 32;
        int blocks = (int)((threads + 255) / 256);
        gcn_scatter<<<blocks, 256, 0, stream>>>(erow, ecol, ev, S, out, E);
    }
}